// HierarchicalMatch_87909390615126
// MI455X (gfx1250) — compile-verified
//
#include <hip/hip_runtime.h>

// HierarchicalMatch on MI455X (gfx1250, wave32).
//
// Shapes: B=128, T=128, D=512, N=3 (9 feature tensors f_ij of [B,T,D] fp32).
// The computation collapses to:
//   fmean[p,b,d] = mean_t f_p[b,t,d]                (p = 3*i+j)   <-- 302 MB stream, dominant
//   num[i,j,b]   = <joint[b], fmean[i,j,b]>
//   bw[i,j,b]    = num / (sum_j num + eps)
//   fii[i,b,d]   = sum_j bw[i,j,b] * fmean[i,j,b,d]
//   num2[i,b]    = <joint[b], fii[i,b]>
//   lam[i,b]     = num2 / (sum_i num2 + eps)
//   out[b,d]     = sum_i lam[i,b] * fii[i,b,d]
//
// Roofline: 302 MB / 23.3 TB/s ~= 13 us; compute is free. Stage 1 does the
// T-reduction on the matrix pipe via V_WMMA_F32_16X16X4_F32 with B = ones:
// C[m,n] = sum_k A[m,k] (row sums). Data goes in A (documented 16x4 fp32
// layout: M = lane&15, K split across the 2 A VGPRs / lane halves); with
// B = ones the result is invariant to the K-slot permutation, so only the
// M<->d mapping must be exact.

typedef __attribute__((ext_vector_type(2))) float v2f;
typedef __attribute__((ext_vector_type(8))) float v8f;

#define B_N   128
#define T_N   128
#define D_N   512
#define NPAIR 9
#define EPSF  1e-8f

struct FPtrs { const float* p[NPAIR]; };

// ---------------------------------------------------------------------------
// Stage 1: fmean[p*B + b][d] = mean over T of f_p[b, t, d], via WMMA row-sums.
// One wave per (pair, b, 16-wide d-tile): 9*128*32 = 36864 waves.
// ---------------------------------------------------------------------------
__global__ __launch_bounds__(256) void fmean_wmma_kernel(FPtrs fp, float* __restrict__ fmean) {
    const int lane  = threadIdx.x & 31;
    const int wave  = blockIdx.x * 8 + (threadIdx.x >> 5);

    const int dTile = wave & 31;        // 0..31  -> d0 = dTile*16
    const int rest  = wave >> 5;
    const int b     = rest & 127;       // 0..127
    const int pair  = rest >> 7;        // 0..8

    const int m    = lane & 15;         // A-matrix row M  == d offset in tile
    const int half = lane >> 4;         // lane half: K = 2*half + {0,1} per ISA A-layout
    const int d0   = dTile * 16;

    // Column (d0+m) of f_p[b], rows stride D. Lane half picks rows t0+2h, t0+2h+1.
    const float* base = fp.p[pair]
                      + (size_t)b * (T_N * D_N)
                      + (size_t)(2 * half) * D_N
                      + d0 + m;

    v2f ones; ones.x = 1.0f; ones.y = 1.0f;   // B matrix = all ones (4x16)
    v8f acc = {};

    // 32 chained WMMAs cover T=128 in K=4 chunks. EXEC is all-ones here (no
    // divergence inside the loop) as WMMA requires.
    #pragma unroll 8
    for (int t0 = 0; t0 < T_N; t0 += 4) {
        const float* p = base + (size_t)t0 * D_N;
        v2f a;
        a.x = p[0];        // A[m, 2*half]   = f[b, t0+2h,   d0+m]
        a.y = p[D_N];      // A[m, 2*half+1] = f[b, t0+2h+1, d0+m]
        // (neg_a, A, neg_b, B, c_mod, C, reuse_a, reuse_b)
        acc = __builtin_amdgcn_wmma_f32_16x16x4_f32(
            false, a, false, ones, (short)0, acc, false, false);
    }

    // C layout: lane 0 holds C[M=0..7, N=0] in acc[0..7] -> sums for d0+0..7;
    // lane 16 holds C[M=8..15, N=0]          -> sums for d0+8..15.
    const float s = 1.0f / (float)T_N;
    const size_t outBase = ((size_t)(pair * B_N + b)) * D_N + d0;
    if (lane == 0) {
        #pragma unroll
        for (int v = 0; v < 8; ++v) fmean[outBase + v] = acc[v] * s;
    } else if (lane == 16) {
        #pragma unroll
        for (int v = 0; v < 8; ++v) fmean[outBase + 8 + v] = acc[v] * s;
    }
}

// ---------------------------------------------------------------------------
// Stage 2: per-b weighting chain. One 256-thread block per b; each thread owns
// d = tid and d = tid+256. Two shuffle+LDS reduction rounds (9 dots, then 3).
// ---------------------------------------------------------------------------
__device__ inline float waveSum(float v) {
    #pragma unroll
    for (int o = 16; o > 0; o >>= 1) v += __shfl_down(v, o, 32);
    return v;
}

__global__ __launch_bounds__(256) void combine_kernel(const float* __restrict__ joint,
                                                      const float* __restrict__ fmean,
                                                      float* __restrict__ out) {
    const int b    = blockIdx.x;
    const int tid  = threadIdx.x;
    const int lane = tid & 31;
    const int wid  = tid >> 5;
    const int d0   = tid;
    const int d1   = tid + 256;

    const float j0 = joint[(size_t)b * D_N + d0];
    const float j1 = joint[(size_t)b * D_N + d1];

    float fm0[NPAIR], fm1[NPAIR];
    #pragma unroll
    for (int k = 0; k < NPAIR; ++k) {
        const size_t base = ((size_t)(k * B_N + b)) * D_N;
        fm0[k] = fmean[base + d0];
        fm1[k] = fmean[base + d1];
    }

    __shared__ float red9[NPAIR][8];
    __shared__ float num9[NPAIR];
    #pragma unroll
    for (int k = 0; k < NPAIR; ++k) {
        float r = waveSum(j0 * fm0[k] + j1 * fm1[k]);
        if (lane == 0) red9[k][wid] = r;
    }
    __syncthreads();
    if (tid < NPAIR) {
        float s = 0.0f;
        #pragma unroll
        for (int w = 0; w < 8; ++w) s += red9[tid][w];
        num9[tid] = s;
    }
    __syncthreads();

    // bw[i][j] = num[i][j] / (sum_j num[i][j] + eps); fii = sum_j bw*fmean
    float fii0[3], fii1[3];
    #pragma unroll
    for (int i = 0; i < 3; ++i) {
        const float den = num9[3 * i] + num9[3 * i + 1] + num9[3 * i + 2] + EPSF;
        const float w0 = num9[3 * i]     / den;
        const float w1 = num9[3 * i + 1] / den;
        const float w2 = num9[3 * i + 2] / den;
        fii0[i] = w0 * fm0[3 * i] + w1 * fm0[3 * i + 1] + w2 * fm0[3 * i + 2];
        fii1[i] = w0 * fm1[3 * i] + w1 * fm1[3 * i + 1] + w2 * fm1[3 * i + 2];
    }

    __shared__ float red3[3][8];
    __shared__ float num3[3];
    #pragma unroll
    for (int i = 0; i < 3; ++i) {
        float r = waveSum(j0 * fii0[i] + j1 * fii1[i]);
        if (lane == 0) red3[i][wid] = r;
    }
    __syncthreads();
    if (tid < 3) {
        float s = 0.0f;
        #pragma unroll
        for (int w = 0; w < 8; ++w) s += red3[tid][w];
        num3[tid] = s;
    }
    __syncthreads();

    const float den2 = num3[0] + num3[1] + num3[2] + EPSF;
    const float l0 = num3[0] / den2;
    const float l1 = num3[1] / den2;
    const float l2 = num3[2] / den2;

    out[(size_t)b * D_N + d0] = l0 * fii0[0] + l1 * fii0[1] + l2 * fii0[2];
    out[(size_t)b * D_N + d1] = l0 * fii1[0] + l1 * fii1[1] + l2 * fii1[2];
}

// ---------------------------------------------------------------------------
extern "C" void kernel_launch(void* const* d_in, const int* in_sizes, int n_in,
                              void* d_out, int out_size, void* d_ws, size_t ws_size,
                              hipStream_t stream) {
    (void)in_sizes; (void)n_in; (void)out_size; (void)ws_size;

    const float* joint = (const float*)d_in[0];
    FPtrs fp;
    for (int k = 0; k < NPAIR; ++k) fp.p[k] = (const float*)d_in[1 + k];

    float* fmean = (float*)d_ws;        // 9*128*512 fp32 = 2.36 MB scratch
    float* outp  = (float*)d_out;       // [128, 512] fp32

    // 9 pairs * 128 b * 32 d-tiles = 36864 waves; 8 waves (256 thr) per block.
    fmean_wmma_kernel<<<dim3(4608), dim3(256), 0, stream>>>(fp, fmean);
    combine_kernel<<<dim3(B_N), dim3(256), 0, stream>>>(joint, fmean, outp);
}